// GraphNeuralNetwork_54056458388016
// MI455X (gfx1250) — compile-verified
//
#include <hip/hip_runtime.h>
#include <cstdint>
#include <cstddef>

// ---------------------------------------------------------------------------
// GraphConv GNN for MI455X (gfx1250, wave32).
//   per layer: agg = scatter_add(x[src] -> dst);  x = agg@Wrel^T + x@Wroot^T + b
//   out = x @ fc_w^T + fc_b
// Scatter phase: L2-resident gathers + global_atomic_add_f32 (memory bound).
// GEMM phase: V_WMMA_F32_16X16X4_F32, register-blocked over all n-tiles so
// each K-step issues NT independent WMMA chains (ILP for the matrix pipe).
// ---------------------------------------------------------------------------

typedef float v2f __attribute__((ext_vector_type(2)));
typedef float v8f __attribute__((ext_vector_type(8)));

#define D_FEAT 128

// ---------------------------------------------------------------- zero agg
__global__ void zero_f32(float* __restrict__ p, size_t n4 /* float4 count */) {
    size_t i      = (size_t)blockIdx.x * blockDim.x + threadIdx.x;
    size_t stride = (size_t)gridDim.x * blockDim.x;
    const float4 z = make_float4(0.f, 0.f, 0.f, 0.f);
    for (; i < n4; i += stride)
        reinterpret_cast<float4*>(p)[i] = z;
}

// ------------------------------------------------------- edge scatter-add
// One wave per edge: 32 lanes x float4 = full 128-float row. Gather is one
// coalesced 512B row read (node table is 51MB < 192MB L2, so L2-resident);
// scatter is 4 native f32 atomics per lane (no-return -> STOREcnt path).
__global__ void scatter_add_edges(const float* __restrict__ x,
                                  const long long* __restrict__ src,
                                  const long long* __restrict__ dst,
                                  float* __restrict__ agg, int n_edges) {
    const int lane    = threadIdx.x & 31;
    const int wave    = blockIdx.x * (blockDim.x >> 5) + (threadIdx.x >> 5);
    const int n_waves = gridDim.x * (blockDim.x >> 5);
    for (int e = wave; e < n_edges; e += n_waves) {
        const long long s = src[e];   // wave-uniform, HW broadcasts
        const long long d = dst[e];
        const float4 v =
            *reinterpret_cast<const float4*>(x + (size_t)s * D_FEAT + lane * 4);
        float* o = agg + (size_t)d * D_FEAT + lane * 4;
        __hip_atomic_fetch_add(o + 0, v.x, __ATOMIC_RELAXED, __HIP_MEMORY_SCOPE_AGENT);
        __hip_atomic_fetch_add(o + 1, v.y, __ATOMIC_RELAXED, __HIP_MEMORY_SCOPE_AGENT);
        __hip_atomic_fetch_add(o + 2, v.z, __ATOMIC_RELAXED, __HIP_MEMORY_SCOPE_AGENT);
        __hip_atomic_fetch_add(o + 3, v.w, __ATOMIC_RELAXED, __HIP_MEMORY_SCOPE_AGENT);
    }
}

// --------------------------------------------------------- fused WMMA GEMM
// out[m, n] = sum_k A1[m,k]*W1[n,k] (+ A2[m,k]*W2[n,k]) + bias[n]
// A: [n_rows, 128]; W: [NT*16, 128]; out: [n_rows, NT*16].
//
// f32 16x4 A-fragment (ISA 7.12.2): lane L<16 -> row m0+L, K = k0..k0+1;
// lane L>=16 -> row m0+L-16, K = k0+2..k0+3 -> one b64 load per lane at a
// fixed per-lane base + immediate offset. B = W^T has the identical load
// pattern with "row" = output column of W (W stored [out,in] row-major).
//
// One wave per 16-row M-tile; all NT accumulators (NT*8 VGPRs) live in
// registers across the K-loop. Per K-step: A-frags loaded once and reused by
// NT independent WMMAs -> independent accumulation chains, no spills.
template <bool DUAL, int NT>
__global__ __launch_bounds__(256, 1)
void gemm_wmma_f32(const float* __restrict__ A1, const float* __restrict__ W1,
                   const float* __restrict__ A2, const float* __restrict__ W2,
                   const float* __restrict__ bias, float* __restrict__ out,
                   int n_rows) {
    const int lane = threadIdx.x & 31;
    const int wave = threadIdx.x >> 5;
    const int m0   = (blockIdx.x * (blockDim.x >> 5) + wave) * 16;
    if (m0 >= n_rows) return;         // wave-uniform: EXEC all-ones for WMMA

    const int lo    = lane & 15;      // row/col within tile
    const int kbase = (lane >> 4) << 1;

    // Per-lane fragment base pointers; every frag load = base + imm offset.
    const float* pa1 = A1 + (size_t)(m0 + lo) * D_FEAT + kbase;
    const float* pa2 = DUAL ? (A2 + (size_t)(m0 + lo) * D_FEAT + kbase) : nullptr;
    const float* pb1 = W1 + (size_t)lo * D_FEAT + kbase;
    const float* pb2 = DUAL ? (W2 + (size_t)lo * D_FEAT + kbase) : nullptr;

    v8f c[NT];
#pragma unroll
    for (int t = 0; t < NT; ++t) c[t] = (v8f){};

#pragma unroll 4
    for (int kk = 0; kk < D_FEAT / 4; ++kk) {
        const v2f a1 = *reinterpret_cast<const v2f*>(pa1 + kk * 4);
        v2f a2;
        if (DUAL) a2 = *reinterpret_cast<const v2f*>(pa2 + kk * 4);
#pragma unroll
        for (int t = 0; t < NT; ++t) {
            const v2f b1 =
                *reinterpret_cast<const v2f*>(pb1 + t * 16 * D_FEAT + kk * 4);
            c[t] = __builtin_amdgcn_wmma_f32_16x16x4_f32(
                       false, a1, false, b1, (short)0, c[t], false, false);
            if (DUAL) {
                const v2f b2 =
                    *reinterpret_cast<const v2f*>(pb2 + t * 16 * D_FEAT + kk * 4);
                c[t] = __builtin_amdgcn_wmma_f32_16x16x4_f32(
                           false, a2, false, b2, (short)0, c[t], false, false);
            }
        }
    }

    // C/D layout: vgpr r -> row m0 + r + 8*(lane>=16), col n0 + (lane&15)
    const int n_cols = NT * 16;
    const int rbase  = m0 + ((lane >> 4) << 3);
#pragma unroll
    for (int t = 0; t < NT; ++t) {
        const int   col = t * 16 + lo;
        const float bv  = bias[col];
#pragma unroll
        for (int r = 0; r < 8; ++r)
            out[(size_t)(rbase + r) * n_cols + col] = c[t][r] + bv;
    }
}

// ---------------------------------------------------------------- launcher
extern "C" void kernel_launch(void* const* d_in, const int* in_sizes, int n_in,
                              void* d_out, int out_size, void* d_ws, size_t ws_size,
                              hipStream_t stream) {
    (void)n_in; (void)ws_size;
    const float*     x      = (const float*)d_in[0];      // [N,128] f32
    const long long* eidx   = (const long long*)d_in[1];  // [2,E]   i64
    const float*     W_rel  = (const float*)d_in[2];      // [L,128,128]
    const float*     b_rel  = (const float*)d_in[3];      // [L,128]
    const float*     W_root = (const float*)d_in[4];      // [L,128,128]
    const float*     fc_w   = (const float*)d_in[5];      // [64,128]
    const float*     fc_b   = (const float*)d_in[6];      // [64]
    float*           out    = (float*)d_out;              // [N,64]

    const int N = in_sizes[0] / D_FEAT;
    const int E = in_sizes[1] / 2;
    const int L = in_sizes[3] / D_FEAT;

    const size_t feat_elems = (size_t)N * D_FEAT;
    float* buf0 = (float*)d_ws;                // x double-buffer
    float* buf1 = buf0 + feat_elems;
    float* agg  = buf1 + feat_elems;           // neighbor-sum buffer

    const long long* src = eidx;
    const long long* dst = eidx + E;

    const dim3 blk(256);
    const int gemm_blocks = (N + 127) / 128;   // 8 waves x 16 rows per block
    const int scat_blocks = 8192;
    const int zero_blocks = 2048;

    const float* xc = x;
    for (int l = 0; l < L; ++l) {
        zero_f32<<<zero_blocks, blk, 0, stream>>>(agg, feat_elems / 4);
        scatter_add_edges<<<scat_blocks, blk, 0, stream>>>(xc, src, dst, agg, E);
        float* xn = (l & 1) ? buf1 : buf0;
        gemm_wmma_f32<true, D_FEAT / 16><<<gemm_blocks, blk, 0, stream>>>(
            agg, W_rel + (size_t)l * D_FEAT * D_FEAT,
            xc,  W_root + (size_t)l * D_FEAT * D_FEAT,
            b_rel + (size_t)l * D_FEAT, xn, N);
        xc = xn;
    }
    // Final head: [N,128] @ [64,128]^T + fc_b  (out_size/N == 64 columns)
    gemm_wmma_f32<false, 4><<<gemm_blocks, blk, 0, stream>>>(
        xc, fc_w, nullptr, nullptr, fc_b, out, N);
}